// RankingLoss_32195074851283
// MI455X (gfx1250) — compile-verified
//
#include <hip/hip_runtime.h>
#include <hip/hip_bf16.h>

typedef __attribute__((ext_vector_type(2))) float v2f;
typedef __attribute__((ext_vector_type(8))) float v8f;

#define NCLS 10
#define NBLK 1024
#define NTHR 256

// Exact f32 wave(32) sum using V_WMMA_F32_16X16X4_F32.
// A per-lane = {v, 0}  => A row m = {v_m, 0, v_{m+16}, 0}
// B = ones(4x16), C = 0 => D[m][n] = v_m + v_{m+16}
// Lane L<16 holds D[0..7][L]  -> sum_{m=0..7} (v_m+v_{m+16})
// Lane L>=16 holds D[8..15][L-16] -> sum_{m=8..15}(v_m+v_{m+16})
// readlane(0) + readlane(16) = full 32-lane sum (uniform result).
__device__ __forceinline__ float wave_sum_wmma(float v) {
    v2f a; a[0] = v;    a[1] = 0.0f;
    v2f b; b[0] = 1.0f; b[1] = 1.0f;
    v8f c = {};
    v8f d = __builtin_amdgcn_wmma_f32_16x16x4_f32(
        /*neg_a=*/false, a, /*neg_b=*/false, b,
        /*c_mod=*/(short)0, c, /*reuse_a=*/false, /*reuse_b=*/false);
    float s = ((d[0] + d[1]) + (d[2] + d[3])) + ((d[4] + d[5]) + (d[6] + d[7]));
    int si = __builtin_bit_cast(int, s);
    float lo = __builtin_bit_cast(float, __builtin_amdgcn_readlane(si, 0));
    float hi = __builtin_bit_cast(float, __builtin_amdgcn_readlane(si, 16));
    return lo + hi;
}

// Kernel 1: ranking loss depends only on target class t (10 values).
// f(t) = sum_{j != t} max(0, 1 + r[j] - r[t]) where r = argsort(distance_matrix[t]).
__global__ void rank_table_kernel(const float* __restrict__ dm,
                                  float* __restrict__ ftab) {
    int t = threadIdx.x;
    if (t >= NCLS) return;
    float dv[NCLS];
#pragma unroll
    for (int j = 0; j < NCLS; ++j) dv[j] = dm[t * NCLS + j];

    // Stable sort position of element j (matches jnp.argsort stability).
    int pos[NCLS];
#pragma unroll
    for (int j = 0; j < NCLS; ++j) {
        int p = 0;
#pragma unroll
        for (int k = 0; k < NCLS; ++k)
            p += (dv[k] < dv[j]) || ((dv[k] == dv[j]) && (k < j));
        pos[j] = p;
    }
    // r[q] = index of q-th smallest (argsort output), all-static indexing.
    float r[NCLS];
#pragma unroll
    for (int q = 0; q < NCLS; ++q) {
        int idx = 0;
#pragma unroll
        for (int k = 0; k < NCLS; ++k) idx += (pos[k] == q) ? k : 0;
        r[q] = (float)idx;
    }
    float rt = 0.0f;
#pragma unroll
    for (int q = 0; q < NCLS; ++q) rt = (q == t) ? r[q] : rt;
    float loss = 0.0f;
#pragma unroll
    for (int j = 0; j < NCLS; ++j)
        loss += (j == t) ? 0.0f : fmaxf(0.0f, 1.0f + r[j] - rt);
    ftab[t] = loss;
}

// Kernel 2: streaming LSE + gather + rank-table add; block partial sums.
__global__ void __launch_bounds__(NTHR)
ce_rank_kernel(const float* __restrict__ inp, const int* __restrict__ tgt,
               const float* __restrict__ ftab, float* __restrict__ partials,
               int n) {
    __shared__ float fsh[NCLS];
    __shared__ float wsum[NTHR / 32];
    if (threadIdx.x < NCLS) fsh[threadIdx.x] = ftab[threadIdx.x];
    __syncthreads();

    float acc = 0.0f;
    int stride = (int)(gridDim.x * blockDim.x);
    for (int i = (int)(blockIdx.x * blockDim.x + threadIdx.x); i < n; i += stride) {
        // Row = 40 bytes, 8B-aligned: five NT b64 loads (single-pass stream).
        const v2f* p = (const v2f*)(inp + (size_t)i * NCLS);
        v2f q0 = __builtin_nontemporal_load(p + 0);
        v2f q1 = __builtin_nontemporal_load(p + 1);
        v2f q2 = __builtin_nontemporal_load(p + 2);
        v2f q3 = __builtin_nontemporal_load(p + 3);
        v2f q4 = __builtin_nontemporal_load(p + 4);
        float x[NCLS] = {q0[0], q0[1], q1[0], q1[1], q2[0],
                         q2[1], q3[0], q3[1], q4[0], q4[1]};
        int t = __builtin_nontemporal_load(tgt + i);

        float m = x[0];
#pragma unroll
        for (int j = 1; j < NCLS; ++j) m = fmaxf(m, x[j]);
        float s = 0.0f, xt = 0.0f;
#pragma unroll
        for (int j = 0; j < NCLS; ++j) {
            s += __expf(x[j] - m);
            xt = (j == t) ? x[j] : xt;
        }
        acc += (m + __logf(s)) - xt + fsh[t];
    }

    float wtot = wave_sum_wmma(acc);   // EXEC all-1s: every thread reaches here
    int lane = threadIdx.x & 31, w = (int)(threadIdx.x >> 5);
    if (lane == 0) wsum[w] = wtot;
    __syncthreads();
    if (threadIdx.x == 0) {
        float tsum = 0.0f;
#pragma unroll
        for (int k = 0; k < NTHR / 32; ++k) tsum += wsum[k];
        partials[blockIdx.x] = tsum;
    }
}

// Kernel 3: deterministic final reduction + divide by B.
__global__ void __launch_bounds__(NTHR)
finalize_kernel(const float* __restrict__ partials, float* __restrict__ out,
                int nPart, float invB) {
    __shared__ float wsum[NTHR / 32];
    float acc = 0.0f;
    for (int i = (int)threadIdx.x; i < nPart; i += NTHR) acc += partials[i];
    float wtot = wave_sum_wmma(acc);
    int lane = threadIdx.x & 31, w = (int)(threadIdx.x >> 5);
    if (lane == 0) wsum[w] = wtot;
    __syncthreads();
    if (threadIdx.x == 0) {
        float t = 0.0f;
#pragma unroll
        for (int k = 0; k < NTHR / 32; ++k) t += wsum[k];
        out[0] = t * invB;
    }
}

extern "C" void kernel_launch(void* const* d_in, const int* in_sizes, int n_in,
                              void* d_out, int out_size, void* d_ws, size_t ws_size,
                              hipStream_t stream) {
    const float* inp = (const float*)d_in[0];   // [B, 10] f32
    const float* dm  = (const float*)d_in[1];   // [10, 10] f32
    const int*   tgt = (const int*)d_in[2];     // [B] int32 (JAX x64-off)
    float* out = (float*)d_out;

    int n = in_sizes[0] / NCLS;                 // B rows
    float* ws = (float*)d_ws;
    float* ftab     = ws;                       // 16 floats reserved
    float* partials = ws + 16;                  // NBLK floats

    rank_table_kernel<<<1, 32, 0, stream>>>(dm, ftab);
    ce_rank_kernel<<<NBLK, NTHR, 0, stream>>>(inp, tgt, ftab, partials, n);
    finalize_kernel<<<1, NTHR, 0, stream>>>(partials, out, NBLK, 1.0f / (float)n);
}